// BuiltinGCN_8443905704047
// MI455X (gfx1250) — compile-verified
//
#include <hip/hip_runtime.h>

typedef float v2f __attribute__((ext_vector_type(2)));
typedef float v4f __attribute__((ext_vector_type(4)));
typedef float v8f __attribute__((ext_vector_type(8)));
typedef unsigned int u32x4 __attribute__((ext_vector_type(4)));
typedef int i32x4 __attribute__((ext_vector_type(4)));
typedef int i32x8 __attribute__((ext_vector_type(8)));

#define HF   1024      // feature dim (IN_FEAT == H_FEAT)
#define NCLS 64        // num classes
#define MP   10240     // node rows padded to multiple of 256

#define MB   256       // block tile M (8 waves x 32 rows)
#define NB   64        // block tile N (4 wmma tiles per wave)
#define KB   32        // K panel staged in LDS
#define LDA_T 36       // A LDS pitch (floats): 32 + 4 TDM pad dwords, conflict-free
#define LDB_T 72       // B LDS pitch (floats): 64 + 8 TDM pad dwords, conflict-free

// ---------------------------------------------------------------- utilities

__global__ void fill_zero_k(float* __restrict__ p, long n) {
    long i = (long)blockIdx.x * blockDim.x + threadIdx.x;
    long stride = (long)gridDim.x * blockDim.x;
    for (; i < n; i += stride) p[i] = 0.0f;
}

__global__ void degree_k(const int* __restrict__ src, const int* __restrict__ dst,
                         float* __restrict__ deg_out, float* __restrict__ deg_in, int E) {
    int e = blockIdx.x * blockDim.x + threadIdx.x;
    if (e < E) {
        atomicAdd(&deg_out[src[e]], 1.0f);
        atomicAdd(&deg_in[dst[e]], 1.0f);
    }
}

__global__ void norm_k(const float* __restrict__ deg, float* __restrict__ nrm, int n) {
    int i = blockIdx.x * blockDim.x + threadIdx.x;
    if (i < n) nrm[i] = rsqrtf(fmaxf(deg[i], 1.0f));
}

// one block (256 threads) per edge; each thread scatters 4 floats
__global__ __launch_bounds__(256) void scatter1024_k(
    const float* __restrict__ x, const int* __restrict__ src, const int* __restrict__ dst,
    const float* __restrict__ nsrc, float* __restrict__ agg) {
    int e = blockIdx.x;
    int s = src[e];
    int d = dst[e];
    float ns = nsrc[s];
    int t = threadIdx.x;
    v4f v = *(const v4f*)(x + (long)s * HF + t * 4);
    float* o = agg + (long)d * HF + t * 4;
    atomicAdd(o + 0, v[0] * ns);
    atomicAdd(o + 1, v[1] * ns);
    atomicAdd(o + 2, v[2] * ns);
    atomicAdd(o + 3, v[3] * ns);
}

__global__ void scatter64_k(const float* __restrict__ t3, const int* __restrict__ src,
                            const int* __restrict__ dst, const float* __restrict__ nsrc,
                            float* __restrict__ oacc, int E) {
    long i = (long)blockIdx.x * blockDim.x + threadIdx.x;
    if (i >= (long)E * NCLS) return;
    int e = (int)(i >> 6);
    int c = (int)(i & 63);
    int s = src[e];
    atomicAdd(&oacc[(long)dst[e] * NCLS + c], t3[(long)s * NCLS + c] * nsrc[s]);
}

__global__ void final_k(const float* __restrict__ oacc, const float* __restrict__ ndst,
                        const float* __restrict__ b3, float* __restrict__ out, int nNodes) {
    long i = (long)blockIdx.x * blockDim.x + threadIdx.x;
    if (i >= (long)nNodes * NCLS) return;
    int m = (int)(i >> 6);
    int c = (int)(i & 63);
    out[i] = oacc[(long)m * NCLS + c] * ndst[m] + b3[c];
}

// ---------------------------------------------------------------- TDM helper
// 2-D TENSOR_LOAD_TO_LDS: tile of tileH rows x tileW f32 elements, global row
// stride strideElems, written to LDS at byte offset ldsByte with padAmount
// dwords appended after every padded interval (pitch = interval + pad).
// tensor_dim == tile_dim (tile coords are relative to global_addr -> no OOB).
__device__ __forceinline__ void tdm_load_2d(const float* gsrc, unsigned ldsByte,
                                            unsigned tileW, unsigned tileH,
                                            unsigned strideElems,
                                            unsigned padIntervalCode,   // 2^(c+1) dw
                                            unsigned padAmountCode) {   // c+1 dw
    unsigned long long ga = (unsigned long long)(uintptr_t)gsrc;
    u32x4 g0;
    g0.x = 1u;                                             // count=1 (valid D#)
    g0.y = ldsByte;                                        // lds_addr
    g0.z = (unsigned)(ga & 0xffffffffull);                 // global_addr[31:0]
    g0.w = (unsigned)((ga >> 32) & 0x01ffffffull)          // global_addr[56:32]
           | (2u << 30);                                   // type = 2 ("image")
    i32x8 g1;
    g1[0] = (int)((2u << 16)                 // data_size = 2 -> 4 bytes
                | (1u << 20)                 // pad_enable
                | (padIntervalCode << 22)
                | (padAmountCode << 25));
    unsigned td0 = tileW, td1 = tileH;
    g1[1] = (int)((td0 & 0xffffu) << 16);                  // [47:32]=0, tensor_dim0.lo
    g1[2] = (int)((td0 >> 16) | ((td1 & 0xffffu) << 16));  // dim0.hi, dim1.lo
    g1[3] = (int)((td1 >> 16) | (tileW << 16));            // dim1.hi, tile_dim0
    g1[4] = (int)(tileH);                                  // tile_dim1, tile_dim2=0
    g1[5] = (int)strideElems;                              // tensor_dim0_stride lo32
    g1[6] = 0;
    g1[7] = 0;
    i32x4 z4 = {0, 0, 0, 0};
#if __has_include(<hip/amd_detail/amd_gfx1250_TDM.h>)
    i32x8 z8 = {0, 0, 0, 0, 0, 0, 0, 0};
    __builtin_amdgcn_tensor_load_to_lds(g0, g1, z4, z4, z8, 0);
#else
    __builtin_amdgcn_tensor_load_to_lds(g0, g1, z4, z4, 0);
#endif
}

// ---------------------------------------------------------------- WMMA GEMM
// C[MP x N] = act( diag(rowscale) * (A @ W) + bias ). fp32 via
// V_WMMA_F32_16X16X4_F32. 8 waves, wave tile 32x64. Double-buffered LDS,
// one barrier per K panel; BOTH A and W panels staged by the Tensor Data
// Mover (wave 0 issues, TENSORcnt-synced) -> compute waves run pure WMMA.
// Row scaling commutes with the matmul, so it is applied in the epilogue.
__global__ __launch_bounds__(256) void gemm_wmma_f32(
    const float* __restrict__ A, const float* __restrict__ W,
    const float* __restrict__ bias, const float* __restrict__ rowscale,
    float* __restrict__ C, int K, int N, int relu) {
    __shared__ float sA[2][MB * LDA_T];   // 2 x 36864 B (TDM pitch 36)
    __shared__ float sB[2][KB * LDB_T];   // 2 x  9216 B (TDM pitch 72)

    const int tid  = threadIdx.x;
    const int wave = tid >> 5;
    const int lane = tid & 31;
    const int half = lane >> 4;
    const int l16  = lane & 15;
    const long mBase = (long)blockIdx.y * MB;
    const int  nBase = blockIdx.x * NB;

    v8f acc[2][4] = {};
    const int np = K / KB;

    // ---- prologue: DMA panel 0
    if (tid < 32) {
        tdm_load_2d(A + mBase * K, (unsigned)(uintptr_t)&sA[0][0],
                    KB, MB, (unsigned)K, 4u, 3u);          // 32 dw rows + 4 pad
        tdm_load_2d(W + nBase, (unsigned)(uintptr_t)&sB[0][0],
                    NB, KB, (unsigned)N, 5u, 7u);          // 64 dw rows + 8 pad
        __builtin_amdgcn_s_wait_tensorcnt(0);
    }
    __syncthreads();

    for (int p = 0; p < np; ++p) {
        const int cur = p & 1;
        const int nxt = cur ^ 1;
        const bool hasNext = (p + 1) < np;

        if (hasNext && tid < 32) {
            const int k0n = (p + 1) * KB;
            tdm_load_2d(A + mBase * K + k0n, (unsigned)(uintptr_t)&sA[nxt][0],
                        KB, MB, (unsigned)K, 4u, 3u);
            tdm_load_2d(W + (long)k0n * N + nBase, (unsigned)(uintptr_t)&sB[nxt][0],
                        NB, KB, (unsigned)N, 5u, 7u);
        }

        // ---- compute panel p (pure LDS + WMMA)
        const float* __restrict__ pA = &sA[cur][0];
        const float* __restrict__ pB = &sB[cur][0];
        const int r0 = wave * 32 + l16;
#pragma unroll
        for (int ks = 0; ks < KB; ks += 4) {
            const int kk = ks + half * 2;
            v2f a0 = *(const v2f*)&pA[r0 * LDA_T + kk];
            v2f a1 = *(const v2f*)&pA[(r0 + 16) * LDA_T + kk];
#pragma unroll
            for (int t = 0; t < 4; ++t) {
                v2f b;
                b[0] = pB[kk * LDB_T + t * 16 + l16];
                b[1] = pB[(kk + 1) * LDB_T + t * 16 + l16];
                acc[0][t] = __builtin_amdgcn_wmma_f32_16x16x4_f32(
                    false, a0, false, b, (short)0, acc[0][t], false, false);
                acc[1][t] = __builtin_amdgcn_wmma_f32_16x16x4_f32(
                    false, a1, false, b, (short)0, acc[1][t], false, false);
            }
        }

        if (hasNext) {
            if (tid < 32) __builtin_amdgcn_s_wait_tensorcnt(0);
            __syncthreads();
        }
    }

    // ---- epilogue: row scale (moved across the matmul) + bias + relu + store
    float rsc[2][8];
#pragma unroll
    for (int s = 0; s < 2; ++s)
#pragma unroll
        for (int r = 0; r < 8; ++r) {
            long row = mBase + wave * 32 + s * 16 + r + half * 8;
            rsc[s][r] = rowscale ? rowscale[row] : 1.0f;
        }
#pragma unroll
    for (int s = 0; s < 2; ++s) {
#pragma unroll
        for (int t = 0; t < 4; ++t) {
            int col = nBase + t * 16 + l16;
            float bv = bias ? bias[col] : 0.0f;
#pragma unroll
            for (int r = 0; r < 8; ++r) {
                long row = mBase + wave * 32 + s * 16 + r + half * 8;
                float v = acc[s][t][r] * rsc[s][r] + bv;
                if (relu) v = fmaxf(v, 0.0f);
                C[row * N + col] = v;
            }
        }
    }
}

// ---------------------------------------------------------------- launcher

extern "C" void kernel_launch(void* const* d_in, const int* in_sizes, int n_in,
                              void* d_out, int out_size, void* d_ws, size_t ws_size,
                              hipStream_t stream) {
    (void)n_in; (void)out_size; (void)ws_size;
    const float* h   = (const float*)d_in[0];
    const int*   src = (const int*)d_in[1];
    const int*   dst = (const int*)d_in[2];
    const float* W1  = (const float*)d_in[3];
    const float* b1  = (const float*)d_in[4];
    const float* W2  = (const float*)d_in[5];
    const float* b2  = (const float*)d_in[6];
    const float* W3  = (const float*)d_in[7];
    const float* b3  = (const float*)d_in[8];
    float* out = (float*)d_out;

    const int E  = in_sizes[1];
    const int Nn = in_sizes[0] / HF;   // 10000

    float* ws   = (float*)d_ws;
    float* dego = ws;  ws += MP;
    float* degi = ws;  ws += MP;
    float* nsrc = ws;  ws += MP;
    float* ndst = ws;  ws += MP;
    float* bufA = ws;  ws += (long)MP * HF;   // aggregation accumulator
    float* bufB = ws;  ws += (long)MP * HF;   // GEMM output / next-layer input
    float* bufT = ws;  ws += (long)MP * NCLS; // layer-3 x @ W3
    float* bufO = ws;  ws += (long)MP * NCLS; // layer-3 aggregation accumulator

    // zero accumulators every call (deterministic; ws is poisoned once)
    fill_zero_k<<<64, 256, 0, stream>>>(dego, 2L * MP);
    fill_zero_k<<<4096, 256, 0, stream>>>(bufA, (long)MP * HF);
    fill_zero_k<<<256, 256, 0, stream>>>(bufO, (long)MP * NCLS);

    degree_k<<<(E + 255) / 256, 256, 0, stream>>>(src, dst, dego, degi, E);
    norm_k<<<(MP + 255) / 256, 256, 0, stream>>>(dego, nsrc, MP);
    norm_k<<<(MP + 255) / 256, 256, 0, stream>>>(degi, ndst, MP);

    dim3 gBig(HF / NB, MP / MB);   // (16, 40)
    dim3 gCls(NCLS / NB, MP / MB); // (1, 40)

    // layer 1
    scatter1024_k<<<E, 256, 0, stream>>>(h, src, dst, nsrc, bufA);
    gemm_wmma_f32<<<gBig, 256, 0, stream>>>(bufA, W1, b1, ndst, bufB, HF, HF, 1);

    // layer 2
    fill_zero_k<<<4096, 256, 0, stream>>>(bufA, (long)MP * HF);
    scatter1024_k<<<E, 256, 0, stream>>>(bufB, src, dst, nsrc, bufA);
    gemm_wmma_f32<<<gBig, 256, 0, stream>>>(bufA, W2, b2, ndst, bufB, HF, HF, 1);

    // layer 3 (GEMM-first: agg(x)@W3 == agg(x@W3) -> 16x less scatter traffic)
    gemm_wmma_f32<<<gCls, 256, 0, stream>>>(bufB, W3, nullptr, nullptr, bufT, HF, NCLS, 0);
    scatter64_k<<<(int)(((long)E * NCLS + 255) / 256), 256, 0, stream>>>(bufT, src, dst, nsrc, bufO, E);
    final_k<<<(Nn * NCLS + 255) / 256, 256, 0, stream>>>(bufO, ndst, b3, out, Nn);
}